// InstanceEmbeddingLoss_60748017435239
// MI455X (gfx1250) — compile-verified
//
#include <hip/hip_runtime.h>
#include <hip/hip_bf16.h>

typedef __attribute__((ext_vector_type(2))) float v2f;
typedef __attribute__((ext_vector_type(8))) float v8f;

#define NBINS 65536
#define EMAXV 64.0f          // error range [-64, 64], bin width = 1/512
#define HW_IN (192 * 192)    // interior pixels per image

// workspace layout (32-bit words)
#define WS_CSUM 0                    // 2*32*16 float
#define WS_CCNT 1024                 // 2*32 float
#define WS_CTR  1088                 // 2*32*16 float
#define WS_HTOT 2112                 // 2*65536 uint
#define WS_HGT  (2112 + 2 * NBINS)   // 2*65536 uint
#define WS_WORDS (2112 + 4 * NBINS)

__global__ void zero_ws_kernel(unsigned* __restrict__ w, int n) {
    for (int i = blockIdx.x * blockDim.x + threadIdx.x; i < n;
         i += gridDim.x * blockDim.x)
        w[i] = 0u;
}

// Per-instance masked sums of the embedding (centers numerator) + pixel counts.
// 144 blocks per image (36864 / 256) so each block maps to exactly one image.
__global__ void center_accum_kernel(const float* __restrict__ emb,
                                    const int* __restrict__ gt,
                                    float* __restrict__ csum,
                                    float* __restrict__ ccnt) {
    __shared__ float ls[32 * 16];
    __shared__ float lc[32];
    const int t = threadIdx.x;
    for (int i = t; i < 512; i += 256) ls[i] = 0.0f;
    if (t < 32) lc[t] = 0.0f;
    __syncthreads();

    const int img = blockIdx.x / 144;
    const int p   = (blockIdx.x % 144) * 256 + t;   // interior pixel id
    const int g   = gt[img * HW_IN + p];
    if (g > 0) {
        const int n = g - 1;
        atomicAdd(&lc[n], 1.0f);
        const float* e = emb + (size_t)img * 16 * HW_IN + p;
#pragma unroll
        for (int c = 0; c < 16; ++c)
            atomicAdd(&ls[n * 16 + c], e[c * HW_IN]);
    }
    __syncthreads();
    for (int i = t; i < 512; i += 256)
        if (ls[i] != 0.0f) atomicAdd(&csum[img * 512 + i], ls[i]);
    if (t < 32 && lc[t] != 0.0f) atomicAdd(&ccnt[img * 32 + t], lc[t]);
}

__global__ void center_finalize_kernel(const float* __restrict__ csum,
                                       const float* __restrict__ ccnt,
                                       float* __restrict__ ctr) {
    const int i = blockIdx.x * blockDim.x + threadIdx.x;
    if (i < 1024) ctr[i] = csum[i] / (ccnt[i >> 4] + 1e-6f);
}

// Fused: feats construction -> f32 WMMA GEMM (x@W1+b1) -> relu -> @W2+b2
// -> hinge errors -> Lovasz histogram (per-image GT / total counters).
// One wave = 16 padded pixels of one (image, instance) pair.
__global__ void __launch_bounds__(256)
fused_mlp_lovasz_kernel(const float* __restrict__ emb,
                        const int* __restrict__ gt,
                        const float* __restrict__ W1,
                        const float* __restrict__ b1,
                        const float* __restrict__ W2,
                        const float* __restrict__ b2,
                        const float* __restrict__ ctr,
                        unsigned* __restrict__ hTot,
                        unsigned* __restrict__ hGT) {
    const int lane   = threadIdx.x & 31;
    const int wid    = blockIdx.x * 8 + (threadIdx.x >> 5);  // [0, 262144)
    const int img    = wid >> 17;          // / (32*4096)
    const int inst   = (wid >> 12) & 31;
    const int pbase  = (wid & 4095) << 4;  // 16 consecutive padded pixels
    const int half   = lane >> 4;
    const int within = lane & 15;

    // This lane's padded pixel (256x256 grid incl. 32-px zero pad)
    const int p  = pbase + within;
    const int ih = (p >> 8) - 32;
    const int iw = (p & 255) - 32;
    const bool interior = ((unsigned)ih < 192u) && ((unsigned)iw < 192u);
    const int g = interior ? gt[img * HW_IN + ih * 192 + iw] : 0;
    const bool label = (g == inst + 1);

    // A fragments: chunk i holds channels 4i..4i+3; this lane-half supplies
    // internal K = {2*half, 2*half+1}  (matches 16x4 f32 A VGPR layout).
    const float* cbase = ctr + (img * 32 + inst) * 16;
    const float* ebase = emb + (size_t)img * 16 * HW_IN +
                         (interior ? (ih * 192 + iw) : 0);
    v2f a[4];
#pragma unroll
    for (int i = 0; i < 4; ++i) {
        const int ca = 4 * i + 2 * half;
        const float e0 = interior ? ebase[ca * HW_IN] : 0.0f;
        const float e1 = interior ? ebase[(ca + 1) * HW_IN] : 0.0f;
        const float f0 = (cbase[ca] - e0) * 0.03125f;
        const float f1 = (cbase[ca + 1] - e1) * 0.03125f;
        a[i].x = fminf(fmaxf(f0, -2.0f), 2.0f);
        a[i].y = fminf(fmaxf(f1, -2.0f), 2.0f);
    }
    // B fragments from W1 [16 x 32]: columns 0..15 (bA) and 16..31 (bB).
    v2f bA[4], bB[4];
#pragma unroll
    for (int i = 0; i < 4; ++i) {
        const int kb = 4 * i + 2 * half;
        bA[i].x = W1[kb * 32 + within];
        bA[i].y = W1[(kb + 1) * 32 + within];
        bB[i].x = W1[kb * 32 + 16 + within];
        bB[i].y = W1[(kb + 1) * 32 + 16 + within];
    }
    // Fold b1 into accumulator init (C/D layout: N = within for every VGPR).
    const float bias0 = b1[within], bias1 = b1[16 + within];
    v8f c0, c1;
#pragma unroll
    for (int r = 0; r < 8; ++r) { c0[r] = bias0; c1[r] = bias1; }

#pragma unroll
    for (int i = 0; i < 4; ++i) {
        c0 = __builtin_amdgcn_wmma_f32_16x16x4_f32(false, a[i], false, bA[i],
                                                   (short)0, c0, false, false);
        c1 = __builtin_amdgcn_wmma_f32_16x16x4_f32(false, a[i], false, bB[i],
                                                   (short)0, c1, false, false);
    }

    // relu, dot with W2 (16-lane xor reduction), + b2.
    const float w2a = W2[within], w2b = W2[16 + within];
    const float b2v = b2[0];
    float tsum[8];
#pragma unroll
    for (int r = 0; r < 8; ++r) {
        float tv = fmaxf(c0[r], 0.0f) * w2a + fmaxf(c1[r], 0.0f) * w2b;
        tv += __shfl_xor(tv, 1, 16);
        tv += __shfl_xor(tv, 2, 16);
        tv += __shfl_xor(tv, 4, 16);
        tv += __shfl_xor(tv, 8, 16);
        tsum[r] = tv;   // lanes 0-15: logit row r ; lanes 16-31: row 8+r
    }

    // Writer lanes own the matching pixel's label:
    //   lane r      -> row r      (pixel pbase+r)
    //   lane 24 + r -> row 8 + r  (pixel pbase+8+r)
#pragma unroll
    for (int r = 0; r < 8; ++r) {
        if (lane == r || lane == 24 + r) {
            const float logit = tsum[r] + b2v;
            const float e = 1.0f - logit * (label ? 1.0f : -1.0f);
            int bin = (int)floorf((e + EMAXV) * 512.0f);
            bin = bin < 0 ? 0 : (bin > NBINS - 1 ? NBINS - 1 : bin);
            atomicAdd(&hTot[img * NBINS + bin], 1u);
            if (label) atomicAdd(&hGT[img * NBINS + bin], 1u);
        }
    }
}

// Lovasz hinge from the per-image histogram: descending sweep over bins,
// Abel summation  loss = sum_b relu(e_b) * (jac_after_b - jac_before_b).
__global__ void lovasz_reduce_kernel(const unsigned* __restrict__ hTot,
                                     const unsigned* __restrict__ hGT,
                                     float* __restrict__ out) {
    __shared__ float losses[2];
    const int t = threadIdx.x;
    if (t < 2) {
        const unsigned* ht = hTot + t * NBINS;
        const unsigned* hg = hGT + t * NBINS;
        float G = 0.0f;
        for (int b = 0; b < NBINS; ++b) G += (float)hg[b];
        float cumGT = 0.0f, cumNEG = 0.0f, jprev = 0.0f, loss = 0.0f;
        for (int b = NBINS - 1; b >= 0; --b) {
            const unsigned ct = ht[b];
            if (!ct) continue;
            const unsigned cg = hg[b];
            cumGT += (float)cg;
            cumNEG += (float)(ct - cg);
            const float jac = 1.0f - (G - cumGT) / (G + cumNEG);
            const float eb = ((float)b + 0.5f) * (1.0f / 512.0f) - EMAXV;
            loss += fmaxf(eb, 0.0f) * (jac - jprev);
            jprev = jac;
        }
        losses[t] = loss;
    }
    __syncthreads();
    if (t == 0) out[0] = 0.5f * (losses[0] + losses[1]);
}

extern "C" void kernel_launch(void* const* d_in, const int* in_sizes, int n_in,
                              void* d_out, int out_size, void* d_ws,
                              size_t ws_size, hipStream_t stream) {
    (void)in_sizes; (void)n_in; (void)out_size; (void)ws_size;
    const float* emb = (const float*)d_in[0];  // [2,16,192,192]
    const int*   gt  = (const int*)d_in[1];    // [2,192,192]
    const float* W1  = (const float*)d_in[2];  // [16,32]
    const float* b1  = (const float*)d_in[3];  // [32]
    const float* W2  = (const float*)d_in[4];  // [32]
    const float* b2  = (const float*)d_in[5];  // [1]
    float* out = (float*)d_out;

    unsigned* ws = (unsigned*)d_ws;
    float* csum    = (float*)(ws + WS_CSUM);
    float* ccnt    = (float*)(ws + WS_CCNT);
    float* ctr     = (float*)(ws + WS_CTR);
    unsigned* hTot = ws + WS_HTOT;
    unsigned* hGT  = ws + WS_HGT;

    zero_ws_kernel<<<512, 256, 0, stream>>>(ws, WS_WORDS);
    center_accum_kernel<<<288, 256, 0, stream>>>(emb, gt, csum, ccnt);
    center_finalize_kernel<<<4, 256, 0, stream>>>(csum, ccnt, ctr);
    fused_mlp_lovasz_kernel<<<32768, 256, 0, stream>>>(emb, gt, W1, b1, W2,
                                                       b2, ctr, hTot, hGT);
    lovasz_reduce_kernel<<<1, 64, 0, stream>>>(hTot, hGT, out);
}